// HierarchicalSparseAttentionTriton_36283883717458
// MI455X (gfx1250) — compile-verified
//
#include <hip/hip_runtime.h>

typedef __attribute__((ext_vector_type(2))) float v2f;
typedef __attribute__((ext_vector_type(8))) float v8f;

#define B_ 4
#define N_ 4096
#define H_ 16
#define D_ 64
// per-(b,n) row block: H*D = 1024 floats = 4 KiB

__device__ __forceinline__ float pick8(v8f v, int i) {
    float r = v[0];
    r = (i == 1) ? v[1] : r;
    r = (i == 2) ? v[2] : r;
    r = (i == 3) ? v[3] : r;
    r = (i == 4) ? v[4] : r;
    r = (i == 5) ? v[5] : r;
    r = (i == 6) ? v[6] : r;
    r = (i == 7) ? v[7] : r;
    return r;
}

__global__ __launch_bounds__(32) void hier_attn_kernel(
    const float* __restrict__ Qp, const float* __restrict__ Kp,
    const float* __restrict__ Vp, const float* __restrict__ Kc,
    const float* __restrict__ Vc, float* __restrict__ out)
{
    __shared__ float sQ[1024];   // Q_p  tile: 16 heads x 64
    __shared__ float sK[1024];   // K_p  tile
    __shared__ float sC[2048];   // K_c  tiles: child0 then child1
    __shared__ float sW[H_ * 3]; // softmax weights per head

    const int bid  = blockIdx.x;        // b*N + n
    const int lane = threadIdx.x;       // 0..31 (wave32)
    const size_t base  = (size_t)bid * 1024;  // [B,N,H,D] row block
    const size_t base2 = (size_t)bid * 2048;  // [B,2N,H,D]: rows 2n,2n+1 contiguous

    // Prefetch V lines we will stream in phase 3 (global_prefetch_b8).
    __builtin_prefetch(Vp + base  + (size_t)lane * 32, 0, 1);
    __builtin_prefetch(Vc + base2 + (size_t)lane * 32, 0, 1);
    __builtin_prefetch(Vc + base2 + 1024 + (size_t)lane * 32, 0, 1);

    // ---------------- Phase 1: stage Q, Kp, Kc to LDS (coalesced b128) ----
    {
        const float4* gQ = (const float4*)(Qp + base);
        const float4* gK = (const float4*)(Kp + base);
        const float4* gC = (const float4*)(Kc + base2);
        float4* q4 = (float4*)sQ;
        float4* k4 = (float4*)sK;
        float4* c4 = (float4*)sC;
#pragma unroll
        for (int i = 0; i < 8; ++i)  q4[i * 32 + lane] = gQ[i * 32 + lane];
#pragma unroll
        for (int i = 0; i < 8; ++i)  k4[i * 32 + lane] = gK[i * 32 + lane];
#pragma unroll
        for (int i = 0; i < 16; ++i) c4[i * 32 + lane] = gC[i * 32 + lane];
    }
    __syncthreads();  // single wave: lowers to waits/NOP

    // ---------------- Phase 2: per-head 3-way scores via WMMA f32 16x16x4 -
    // A[m][k] = Q[m][k0+k], B[k][n] = K[n][k0+k]  ->  C = Q*K^T, diag = dots.
    // f32 16x4 operand layout: lanes 0-15 hold row m=lane with K={0,1},
    // lanes 16-31 hold row m=lane-16 with K={2,3}.
    const int h    = lane & 15;
    const int half = lane >> 4;
    const float* qrow  = sQ + h * D_ + 2 * half;
    const float* krow  = sK + h * D_ + 2 * half;
    const float* c0row = sC + h * D_ + 2 * half;
    const float* c1row = sC + 1024 + h * D_ + 2 * half;

    v8f accS = {0.f, 0.f, 0.f, 0.f, 0.f, 0.f, 0.f, 0.f};
    v8f acc0 = accS;
    v8f acc1 = accS;
#pragma unroll
    for (int k0 = 0; k0 < D_; k0 += 4) {
        v2f a  = *(const v2f*)(qrow  + k0);
        v2f bs = *(const v2f*)(krow  + k0);
        v2f b0 = *(const v2f*)(c0row + k0);
        v2f b1 = *(const v2f*)(c1row + k0);
        accS = __builtin_amdgcn_wmma_f32_16x16x4_f32(false, a, false, bs,
                                                     (short)0, accS, false, false);
        acc0 = __builtin_amdgcn_wmma_f32_16x16x4_f32(false, a, false, b0,
                                                     (short)0, acc0, false, false);
        acc1 = __builtin_amdgcn_wmma_f32_16x16x4_f32(false, a, false, b1,
                                                     (short)0, acc1, false, false);
    }

    // Diagonal C[h][h]: lanes 0-7 -> head=lane at acc[lane];
    //                   lanes 24-31 -> head=lane-16 at acc[lane-24].
    const bool isDiag = (lane < 8) || (lane >= 24);
    const int  di     = (lane < 8) ? lane : (lane - 24);
    if (isDiag) {
        const float scale = 0.125f;  // 1/sqrt(64)
        float a0 = pick8(accS, di) * scale;
        float a1 = pick8(acc0, di) * scale;
        float a2 = pick8(acc1, di) * scale;
        float m  = fmaxf(a0, fmaxf(a1, a2));
        float e0 = __expf(a0 - m);
        float e1 = __expf(a1 - m);
        float e2 = __expf(a2 - m);
        float inv = 1.0f / (e0 + e1 + e2 + 1e-9f);
        const int hh = (lane < 8) ? lane : (lane - 16);
        sW[hh * 3 + 0] = e0 * inv;
        sW[hh * 3 + 1] = e1 * inv;
        sW[hh * 3 + 2] = e2 * inv;
    }
    __syncthreads();

    // ---------------- Phase 3: stream V, blend, store (b128) --------------
    {
        const float4* gVp = (const float4*)(Vp + base);
        const float4* gVc = (const float4*)(Vc + base2);
        float4* gO = (float4*)(out + base);
#pragma unroll
        for (int i = 0; i < 8; ++i) {
            const int e4 = i * 32 + lane;   // float4 index within 1024-elem row
            const int hh = e4 >> 4;         // head of this float4
            const float w0 = sW[hh * 3 + 0];
            const float w1 = sW[hh * 3 + 1];
            const float w2 = sW[hh * 3 + 2];
            float4 vp = gVp[e4];
            float4 v0 = gVc[e4];
            float4 v1 = gVc[256 + e4];
            float4 o;
            o.x = fmaf(w0, vp.x, fmaf(w1, v0.x, w2 * v1.x));
            o.y = fmaf(w0, vp.y, fmaf(w1, v0.y, w2 * v1.y));
            o.z = fmaf(w0, vp.z, fmaf(w1, v0.z, w2 * v1.z));
            o.w = fmaf(w0, vp.w, fmaf(w1, v0.w, w2 * v1.w));
            gO[e4] = o;
        }
    }
}

extern "C" void kernel_launch(void* const* d_in, const int* in_sizes, int n_in,
                              void* d_out, int out_size, void* d_ws, size_t ws_size,
                              hipStream_t stream) {
    const float* Qp = (const float*)d_in[0];
    const float* Kp = (const float*)d_in[1];
    const float* Vp = (const float*)d_in[2];
    const float* Kc = (const float*)d_in[3];
    const float* Vc = (const float*)d_in[4];
    float* out = (float*)d_out;
    (void)in_sizes; (void)n_in; (void)out_size; (void)d_ws; (void)ws_size;

    dim3 grid(B_ * N_);   // one wave32 workgroup per (b, n)
    dim3 block(32);
    hier_attn_kernel<<<grid, block, 0, stream>>>(Qp, Kp, Vp, Kc, Vc, out);
}